// HyDRAUniverse_82686710383293
// MI455X (gfx1250) — compile-verified
//
#include <hip/hip_runtime.h>
#include <hip/hip_bf16.h>
#include <math.h>

// ---------------------------------------------------------------------------
// HyDRA pairwise hyperbolic force kernel for gfx1250 (MI455X, wave32, WMMA).
//
// Fused O(N^2) pass, no NxN intermediates. Per 16x16 pair tile:
//   inn tile  : 5x v_wmma_f32_16x16x4_f32  (K=17 padded to 20, A = metric*h_I)
//   physics   : stable acosh + LJ coupling + horizon sigmoid in f32 VALU
//   F spatial : 4x v_wmma_f32_16x16x4_f32  (A = C tile via LDS transpose,
//               accumulator held in WMMA C/D registers across the J loop)
//   row terms : per-lane partials, 16-lane shfl_xor reductions at the end
// Outputs accumulated with global f64 atomics; norm clip in a final kernel.
// ---------------------------------------------------------------------------

#define NPTS   4096
#define NDIM   17
#define ROWP   20        // padded f32 row stride (17 -> 20, zeros above 16)
#define NTILE  256       // 4096 / 16
#define NSPLIT 4         // j-range splits per i-tile

typedef float v2f __attribute__((ext_vector_type(2)));
typedef float v8f __attribute__((ext_vector_type(8)));

// --- prep: f64 h -> padded f32 h32, and g32 = metric * h (time comp negated)
__global__ void hydra_prep(const double* __restrict__ h,
                           float* __restrict__ h32, float* __restrict__ g32) {
    int i = blockIdx.x * blockDim.x + threadIdx.x;
    if (i >= NPTS) return;
    #pragma unroll
    for (int d = 0; d < ROWP; ++d) {
        float v = (d < NDIM) ? (float)h[(size_t)i * NDIM + d] : 0.0f;
        h32[i * ROWP + d] = v;
        g32[i * ROWP + d] = (d == 0) ? -v : v;
    }
}

__global__ void hydra_zero(double* __restrict__ out, int n) {
    int i = blockIdx.x * blockDim.x + threadIdx.x;
    if (i < n) out[i] = 0.0;
}

// --- main fused kernel: one wave per block; block (I, split) does i-tile I,
//     j-tiles J = split, split+NSPLIT, ...
__global__ __launch_bounds__(32)
void hydra_main(const float* __restrict__ h32, const float* __restrict__ g32,
                double* __restrict__ out) {
    __shared__ float csc[16 * 16];           // wave-private C-tile transpose

    const int lane = threadIdx.x;            // 0..31 (wave32)
    const int n0   = lane & 15;              // column-within-tile (N index)
    const int H    = lane >> 4;              // half-wave id
    const int I    = blockIdx.x;
    const int i0   = I * 16;

    // A operand (16x4 f32 layout): lanes 0-15 -> K=4kc+{0,1}, lanes 16-31 -> K=4kc+{2,3}
    v2f aG[5];
    #pragma unroll
    for (int kc = 0; kc < 5; ++kc) {
        const float* gr = g32 + (size_t)(i0 + n0) * ROWP + 4 * kc + 2 * H;
        aG[kc].x = gr[0];
        aG[kc].y = gr[1];
    }

    v8f   Facc = {};                         // F spatial dims 1..16, f32, WMMA acc
    float S0[8], S1[8];                      // row partials: C*h_j0, C*inn
    #pragma unroll
    for (int r = 0; r < 8; ++r) { S0[r] = 0.0f; S1[r] = 0.0f; }
    double Ew = 0.0;

    for (int J = blockIdx.y; J < NTILE; J += NSPLIT) {
        const int j0 = J * 16;

        // hint the next tile's rows into cache (global_prefetch_b8)
        int Jn = J + NSPLIT;
        if (Jn < NTILE)
            __builtin_prefetch(h32 + (size_t)(Jn * 16 + n0) * ROWP, 0, 3);

        // ---- Gram tile: inn = (metric*h_I) . h_J^T via 5 chained WMMAs ----
        v8f acc = {};
        #pragma unroll
        for (int kc = 0; kc < 5; ++kc) {
            const float* hr = h32 + (size_t)(j0 + n0) * ROWP + 4 * kc + 2 * H;
            v2f b; b.x = hr[0]; b.y = hr[1];
            acc = __builtin_amdgcn_wmma_f32_16x16x4_f32(
                false, aG[kc], false, b, (short)0, acc, false, false);
        }

        const float hj0 = h32[(size_t)(j0 + n0) * ROWP];   // time comp of col j
        const int   j   = j0 + n0;

        // ---- elementwise physics on the 8 C/D-layout entries per lane ----
        float Cv[8];
        #pragma unroll
        for (int r = 0; r < 8; ++r) {
            const float innv = acc[r];                     // <h_i, h_j>_L
            const int   i    = i0 + r + 8 * H;
            // stable acosh: d = log1p(t + sqrt(t*(t+2))), t = clamp(-inn,1+eps)-1
            const float x  = fmaxf(-innv, 1.0f + 1e-7f);
            const float t  = x - 1.0f;
            const float Dg = log1pf(t + sqrtf(t * (t + 2.0f)));
            const bool  dg = (i == j);
            // pairwise_dist semantics: zero diag, add D_MIN if below floor
            const float Dp = dg ? 0.0f : (Dg < 0.08f ? Dg + 0.08f : Dg);
            const float De = fmaxf(Dp, 0.05f);
            // dV/dd = exp(-d/SR)/SR - exp(-d/SA)/SA, SR=0.5 SA=2.0
            const float dV = 2.0f * __expf(-2.0f * De) - 0.5f * __expf(-0.5f * De);
            // causal horizon sigmoid((6 - d)/0.5)
            const float sg = 1.0f / (1.0f + __expf((De - 6.0f) * 2.0f));
            const float Kg = dg ? 0.0f : (-dV * sg);
            const float dn = sqrtf(fmaxf(innv * innv - 1.0f, 1e-15f));
            const float C  = Kg * Dg / dn;
            Cv[r] = C;
            S0[r] += C * hj0;                              // time-dim matvec
            S1[r] += C * innv;                             // rowsum(C .* inn)
            // LJ potential on D_pair (diag contributes exactly 0)
            Ew += (double)(__expf(-0.5f * Dp) - __expf(-2.0f * Dp));
        }

        // ---- C tile (C/D layout) -> A layout via wave-private LDS ----
        #pragma unroll
        for (int r = 0; r < 8; ++r)
            csc[(r + 8 * H) * 16 + n0] = Cv[r];
        asm volatile("s_wait_dscnt 0" ::: "memory");       // RAW fence on LDS

        // ---- F spatial: Facc += C(16x16) x h_J[:, 1..16], K=16 as 4 WMMAs ----
        #pragma unroll
        for (int kc = 0; kc < 4; ++kc) {
            const int k = 4 * kc + 2 * H;
            v2f aC; aC.x = csc[n0 * 16 + k]; aC.y = csc[n0 * 16 + k + 1];
            const float* hr = h32 + (size_t)(j0 + k) * ROWP + 1 + n0;
            v2f bH; bH.x = hr[0]; bH.y = hr[ROWP];
            Facc = __builtin_amdgcn_wmma_f32_16x16x4_f32(
                false, aC, false, bH, (short)0, Facc, false, false);
        }
    }

    // ---- reduce row partials across the 16 lanes of each half-wave ----
    #pragma unroll
    for (int r = 0; r < 8; ++r) {
        float s0 = S0[r], s1 = S1[r];
        #pragma unroll
        for (int m = 1; m < 16; m <<= 1) {
            s0 += __shfl_xor(s0, m, 32);
            s1 += __shfl_xor(s1, m, 32);
        }
        S0[r] = s0; S1[r] = s1;
    }
    #pragma unroll
    for (int m = 1; m < 32; m <<= 1) Ew += __shfl_xor(Ew, m, 32);

    // ---- accumulate this split's contribution: F = C@h + rowsum(C.*inn)*h_i
    #pragma unroll
    for (int r = 0; r < 8; ++r) {
        const int i = i0 + r + 8 * H;
        const int d = 1 + n0;
        const double v = (double)Facc[r] +
                         (double)S1[r] * (double)h32[(size_t)i * ROWP + d];
        atomicAdd(&out[(size_t)i * NDIM + d], v);          // global_atomic_add_f64
        if (n0 == 0) {
            const double v0 = (double)S0[r] +
                              (double)S1[r] * (double)h32[(size_t)i * ROWP];
            atomicAdd(&out[(size_t)i * NDIM], v0);
        }
    }
    if (lane == 0) atomicAdd(&out[(size_t)NPTS * NDIM], 0.5 * Ew);
}

// --- finalize: per-row norm clip F *= min(6/||F||, 1)
__global__ void hydra_norm(double* __restrict__ out) {
    int i = blockIdx.x * blockDim.x + threadIdx.x;
    if (i >= NPTS) return;
    double* F = out + (size_t)i * NDIM;
    double s = 0.0;
    #pragma unroll
    for (int d = 0; d < NDIM; ++d) s += F[d] * F[d];
    const double nrm = fmax(sqrt(s), 1e-8);
    const double sc  = fmin(6.0 / nrm, 1.0);
    #pragma unroll
    for (int d = 0; d < NDIM; ++d) F[d] *= sc;
}

extern "C" void kernel_launch(void* const* d_in, const int* in_sizes, int n_in,
                              void* d_out, int out_size, void* d_ws, size_t ws_size,
                              hipStream_t stream) {
    const double* h   = (const double*)d_in[0];          // [4096, 17] f64
    double*       out = (double*)d_out;                  // F [4096,17] + E [1]
    float*        h32 = (float*)d_ws;                    // [4096, 20] f32
    float*        g32 = h32 + (size_t)NPTS * ROWP;       // [4096, 20] f32

    hydra_prep<<<(NPTS + 127) / 128, 128, 0, stream>>>(h, h32, g32);
    hydra_zero<<<(out_size + 255) / 256, 256, 0, stream>>>(out, out_size);

    dim3 grid(NTILE, NSPLIT);
    hydra_main<<<grid, 32, 0, stream>>>(h32, g32, out);

    hydra_norm<<<(NPTS + 127) / 128, 128, 0, stream>>>(out);
}